// ILModel_26723286516182
// MI455X (gfx1250) — compile-verified
//
#include <hip/hip_runtime.h>
#include <math.h>

typedef __attribute__((ext_vector_type(2))) float v2f;
typedef __attribute__((ext_vector_type(8))) float v8f;

#define NODES_ 16384
#define E_     262144
#define F_     256
#define HEADS_ 8
#define HC_    32
#define B_     128
#define N_     128
#define A_     5
#define T_     10
#define HID_   64
#define GRUIN_ 512
#define G3_    192

// ---------------- device helpers ----------------
__device__ __forceinline__ unsigned f2mono(float f) {
  unsigned u = __float_as_uint(f);
  return (u & 0x80000000u) ? ~u : (u | 0x80000000u);
}
__device__ __forceinline__ float mono2f(unsigned u) {
  return __uint_as_float((u & 0x80000000u) ? (u & 0x7fffffffu) : ~u);
}
__device__ __forceinline__ float lrelu(float x) { return x > 0.f ? x : 0.2f * x; }
__device__ __forceinline__ float sigmoidf(float x) { return 1.f / (1.f + expf(-x)); }

// ---------------- utility kernels ----------------
__global__ void kzero(float* __restrict__ p, int n) {
  int i = blockIdx.x * blockDim.x + threadIdx.x;
  if (i < n) p[i] = 0.f;
}
__global__ void kinit_mkey(unsigned* __restrict__ p, int n) {
  int i = blockIdx.x * blockDim.x + threadIdx.x;
  if (i < n) p[i] = 0x007FFFFFu;  // f2mono(-inf)
}
// W (K,N) row-major -> Wt (N,K)
__global__ void ktranspose(const float* __restrict__ W, float* __restrict__ Wt,
                           int K, int N) {
  int i = blockIdx.x * blockDim.x + threadIdx.x;
  if (i >= K * N) return;
  int k = i / N, n = i - k * N;
  Wt[n * K + k] = W[i];
}

// ---------------- WMMA fp32 GEMM: C(MxN) = A(MxK) @ Bt^T, Bt stored (N,K) ----
// Each wave computes a 16x64 strip: 4 accumulator tiles sharing one A fragment.
template <int K>
__global__ __launch_bounds__(128) void kgemmT(const float* __restrict__ A,
                                              const float* __restrict__ Bt,
                                              float* __restrict__ C, int M, int N) {
  int lane = threadIdx.x & 31;
  int tilesN = N >> 6;                               // 64-wide strips
  int total = (M >> 4) * tilesN;
  int tile = blockIdx.x * 4 + (threadIdx.x >> 5);
  if (tile >= total) return;                         // wave-uniform
  int tm = tile / tilesN, tj = tile - tm * tilesN;
  int l15 = lane & 15;
  int koff = (lane >> 4) << 1;                       // 0 or 2
  const float* Ap = A + (size_t)(tm * 16 + l15) * K + koff;
  const float* Bp = Bt + (size_t)(tj * 64 + l15) * K + koff;
  v8f acc0 = {}, acc1 = {}, acc2 = {}, acc3 = {};
#pragma unroll 4
  for (int k = 0; k < K; k += 4) {
    v2f a  = *(const v2f*)(Ap + k);
    v2f b0 = *(const v2f*)(Bp + k);
    v2f b1 = *(const v2f*)(Bp + 16 * K + k);
    v2f b2 = *(const v2f*)(Bp + 32 * K + k);
    v2f b3 = *(const v2f*)(Bp + 48 * K + k);
    acc0 = __builtin_amdgcn_wmma_f32_16x16x4_f32(false, a, false, b0, (short)0, acc0, false, false);
    acc1 = __builtin_amdgcn_wmma_f32_16x16x4_f32(false, a, false, b1, (short)0, acc1, false, false);
    acc2 = __builtin_amdgcn_wmma_f32_16x16x4_f32(false, a, false, b2, (short)0, acc2, false, false);
    acc3 = __builtin_amdgcn_wmma_f32_16x16x4_f32(false, a, false, b3, (short)0, acc3, false, false);
  }
  int row0 = tm * 16 + ((lane >> 4) << 3);
  float* Cp = C + (size_t)row0 * N + tj * 64 + l15;
#pragma unroll
  for (int v = 0; v < 8; ++v) {
    Cp[(size_t)v * N]      = acc0[v];
    Cp[(size_t)v * N + 16] = acc1[v];
    Cp[(size_t)v * N + 32] = acc2[v];
    Cp[(size_t)v * N + 48] = acc3[v];
  }
}

// ---------------- GAT pieces ----------------
__global__ void kce(const float* __restrict__ lin_edge,
                    const float* __restrict__ att_edge, float* __restrict__ ce) {
  int h = threadIdx.x;
  if (h >= HEADS_) return;
  float s = 0.f;
  for (int c = 0; c < HC_; ++c) s += lin_edge[h * HC_ + c] * att_edge[h * HC_ + c];
  ce[h] = s;
}

__global__ void kattn_coef(const float* __restrict__ xp,
                           const float* __restrict__ att_src,
                           const float* __restrict__ att_dst,
                           float* __restrict__ a_src, float* __restrict__ a_dst) {
  int i = blockIdx.x * blockDim.x + threadIdx.x;
  if (i >= NODES_ * HEADS_) return;
  int n = i >> 3, h = i & 7;
  const float* row = xp + n * F_ + h * HC_;
  float s = 0.f, d = 0.f;
  for (int c = 0; c < HC_; ++c) {
    float v = row[c];
    s += v * att_src[h * HC_ + c];
    d += v * att_dst[h * HC_ + c];
  }
  a_src[i] = s;
  a_dst[i] = d;
}

__global__ void kloop_accum(const int* __restrict__ dst, const float* __restrict__ ea,
                            float* __restrict__ sums, float* __restrict__ cnt) {
  int e = blockIdx.x * blockDim.x + threadIdx.x;
  if (e >= E_) return;
  int d = dst[e];
  atomicAdd(&sums[d], ea[e]);
  atomicAdd(&cnt[d], 1.f);
}
__global__ void kloop_attr(const float* __restrict__ sums, const float* __restrict__ cnt,
                           float* __restrict__ la) {
  int i = blockIdx.x * blockDim.x + threadIdx.x;
  if (i < NODES_) la[i] = sums[i] / fmaxf(cnt[i], 1.f);
}

__global__ void kedge_max(const int* __restrict__ src, const int* __restrict__ dst,
                          const float* __restrict__ ea, const float* __restrict__ a_src,
                          const float* __restrict__ a_dst, const float* __restrict__ ce,
                          float* __restrict__ alpha, unsigned* __restrict__ mkey) {
  int i = blockIdx.x * blockDim.x + threadIdx.x;
  if (i >= E_ * HEADS_) return;
  int e = i >> 3, h = i & 7;
  int s = src[e], d = dst[e];
  float a = lrelu(a_src[s * HEADS_ + h] + a_dst[d * HEADS_ + h] + ea[e] * ce[h]);
  alpha[i] = a;
  atomicMax(&mkey[d * HEADS_ + h], f2mono(a));
}
__global__ void kself_max(const float* __restrict__ la, const float* __restrict__ a_src,
                          const float* __restrict__ a_dst, const float* __restrict__ ce,
                          float* __restrict__ alpha_self, unsigned* __restrict__ mkey) {
  int i = blockIdx.x * blockDim.x + threadIdx.x;
  if (i >= NODES_ * HEADS_) return;
  int n = i >> 3, h = i & 7;
  float a = lrelu(a_src[i] + a_dst[i] + la[n] * ce[h]);
  alpha_self[i] = a;
  atomicMax(&mkey[i], f2mono(a));
}

__global__ void kedge_sum(const int* __restrict__ dst, float* __restrict__ alpha,
                          const unsigned* __restrict__ mkey, float* __restrict__ denom) {
  int i = blockIdx.x * blockDim.x + threadIdx.x;
  if (i >= E_ * HEADS_) return;
  int e = i >> 3, h = i & 7;
  int d = dst[e];
  float a = expf(alpha[i] - mono2f(mkey[d * HEADS_ + h]));
  alpha[i] = a;
  atomicAdd(&denom[d * HEADS_ + h], a);
}
__global__ void kself_sum(float* __restrict__ alpha_self,
                          const unsigned* __restrict__ mkey, float* __restrict__ denom) {
  int i = blockIdx.x * blockDim.x + threadIdx.x;
  if (i >= NODES_ * HEADS_) return;
  float a = expf(alpha_self[i] - mono2f(mkey[i]));
  alpha_self[i] = a;
  atomicAdd(&denom[i], a);
}

__global__ void kedge_agg(const int* __restrict__ src, const int* __restrict__ dst,
                          const float* __restrict__ alpha, const float* __restrict__ denom,
                          const float* __restrict__ xp, float* __restrict__ out) {
  int e = blockIdx.x;
  int f = threadIdx.x;           // 256 threads = F_
  int h = f >> 5;
  int s = src[e], d = dst[e];
  float w = alpha[e * HEADS_ + h] / (denom[d * HEADS_ + h] + 1e-16f);
  atomicAdd(&out[d * F_ + f], xp[s * F_ + f] * w);
}
__global__ void kself_agg(const float* __restrict__ alpha_self,
                          const float* __restrict__ denom,
                          const float* __restrict__ xp, float* __restrict__ out) {
  int n = blockIdx.x;
  int f = threadIdx.x;
  int h = f >> 5;
  float w = alpha_self[n * HEADS_ + h] / (denom[n * HEADS_ + h] + 1e-16f);
  atomicAdd(&out[n * F_ + f], xp[n * F_ + f] * w);
}

__global__ void kfinalize(float* __restrict__ out, const float* __restrict__ bias) {
  int i = blockIdx.x * blockDim.x + threadIdx.x;
  if (i >= NODES_ * F_) return;
  out[i] = fmaxf(0.f, out[i] + bias[i & (F_ - 1)]);
}

// ---------------- readout / GRU / head ----------------
__global__ void kpool(const float* __restrict__ y, float* __restrict__ ge) {
  int i = blockIdx.x * blockDim.x + threadIdx.x;
  if (i >= B_ * F_) return;
  int b = i / F_, f = i - b * F_;
  float s = 0.f;
  for (int n = 0; n < N_; ++n) s += y[(b * N_ + n) * F_ + f];
  ge[i] = s * (1.f / (float)N_);
}

__global__ void kcomb(const float* __restrict__ y, const float* __restrict__ ge,
                      float* __restrict__ comb) {
  int i = blockIdx.x * blockDim.x + threadIdx.x;
  if (i >= B_ * A_ * GRUIN_) return;
  int n = i / GRUIN_, col = i - n * GRUIN_;
  int b = n / A_, a = n - b * A_;
  comb[i] = (col < F_) ? y[(b * N_ + a) * F_ + col] : ge[b * F_ + (col - F_)];
}

__global__ void kgru_combine(const float* __restrict__ gi, const float* __restrict__ gh,
                             const float* __restrict__ b_ih, const float* __restrict__ b_hh,
                             float* __restrict__ h) {
  int i = blockIdx.x * blockDim.x + threadIdx.x;
  if (i >= B_ * A_ * HID_) return;
  int n = i >> 6, c = i & 63;
  float ir  = gi[n * G3_ + c]            + b_ih[c];
  float iz  = gi[n * G3_ + HID_ + c]     + b_ih[HID_ + c];
  float in_ = gi[n * G3_ + 2*HID_ + c]   + b_ih[2*HID_ + c];
  float hr  = gh[n * G3_ + c]            + b_hh[c];
  float hz  = gh[n * G3_ + HID_ + c]     + b_hh[HID_ + c];
  float hn  = gh[n * G3_ + 2*HID_ + c]   + b_hh[2*HID_ + c];
  float r = sigmoidf(ir + hr);
  float z = sigmoidf(iz + hz);
  float nc = tanhf(in_ + r * hn);
  float hold = h[i];
  h[i] = (1.f - z) * nc + z * hold;
}

__global__ void khead(const float* __restrict__ h, const float* __restrict__ fc1_w,
                      const float* __restrict__ fc1_b, const float* __restrict__ fc2_w,
                      const float* __restrict__ fc2_b, float* __restrict__ out, int t) {
  int n = blockIdx.x * blockDim.x + threadIdx.x;
  if (n >= B_ * A_) return;
  float hid[HC_];
#pragma unroll 4
  for (int k = 0; k < HC_; ++k) {
    float s = fc1_b[k];
    for (int c = 0; c < HID_; ++c) s += h[n * HID_ + c] * fc1_w[c * HC_ + k];
    hid[k] = fmaxf(0.f, s);
  }
  int b = n / A_, a = n - b * A_;
  for (int o = 0; o < 2; ++o) {
    float s = fc2_b[o];
    for (int k = 0; k < HC_; ++k) s += hid[k] * fc2_w[k * 2 + o];
    out[((b * T_ + t) * A_ + a) * 2 + o] = s;
  }
}

// ---------------- host ----------------
static inline int nb(long n, int t) { return (int)((n + t - 1) / t); }

extern "C" void kernel_launch(void* const* d_in, const int* in_sizes, int n_in,
                              void* d_out, int out_size, void* d_ws, size_t ws_size,
                              hipStream_t stream) {
  (void)in_sizes; (void)n_in; (void)out_size; (void)ws_size;
  const float* x_all  = (const float*)d_in[0];   // (T, NODES, 4)
  const int*   ei_all = (const int*)d_in[1];     // (T, 2, E)
  const float* ea_all = (const float*)d_in[2];   // (T, E, 1)
  int pi = 3;
  struct GatP { const float *W, *asrc, *adst, *lin, *aedge, *bias; } g[3];
  for (int l = 0; l < 3; ++l) {
    g[l].W     = (const float*)d_in[pi++];
    g[l].asrc  = (const float*)d_in[pi++];
    g[l].adst  = (const float*)d_in[pi++];
    g[l].lin   = (const float*)d_in[pi++];
    g[l].aedge = (const float*)d_in[pi++];
    g[l].bias  = (const float*)d_in[pi++];
  }
  const float* w_ih = (const float*)d_in[pi++];  // (192,512) = (N,K)
  const float* w_hh = (const float*)d_in[pi++];  // (192,64)  = (N,K)
  const float* b_ih = (const float*)d_in[pi++];
  const float* b_hh = (const float*)d_in[pi++];
  const float* fc1w = (const float*)d_in[pi++];  // (64,32)
  const float* fc1b = (const float*)d_in[pi++];
  const float* fc2w = (const float*)d_in[pi++];  // (32,2)
  const float* fc2b = (const float*)d_in[pi++];
  float* out = (float*)d_out;                    // (B,T,A,2)

  float* ws = (float*)d_ws;
  size_t o = 0;
  auto take = [&](size_t n) { float* p = ws + o; o += n; return p; };
  float*    bufA   = take((size_t)NODES_ * F_);      // y / aggregation target
  float*    bufB   = take((size_t)NODES_ * F_);      // projected xp
  float*    alpha  = take((size_t)E_ * HEADS_);
  float*    alphaS = take((size_t)NODES_ * HEADS_);
  float*    a_src  = take((size_t)NODES_ * HEADS_);
  float*    a_dst  = take((size_t)NODES_ * HEADS_);
  unsigned* mkey   = (unsigned*)take((size_t)NODES_ * HEADS_);
  float*    denom  = take((size_t)NODES_ * HEADS_);
  float*    la     = take(NODES_);
  float*    sums   = take(NODES_);
  float*    cnt    = take(NODES_);
  float*    ge     = take((size_t)B_ * F_);
  float*    comb   = take((size_t)B_ * A_ * GRUIN_);
  float*    gi     = take((size_t)B_ * A_ * G3_);
  float*    gh     = take((size_t)B_ * A_ * G3_);
  float*    hst    = take((size_t)B_ * A_ * HID_);
  float*    ce     = take(HEADS_);
  float*    Wt1    = take((size_t)F_ * 4);           // (256,4)
  float*    Wt2    = take((size_t)F_ * F_);          // (256,256)
  float*    Wt3    = take((size_t)F_ * F_);          // (256,256)
  float*    Wt[3]  = { Wt1, Wt2, Wt3 };

  const int TB = 256;
  // one-time weight transposes (deterministic, re-done each launch)
  ktranspose<<<nb(4 * F_, TB), TB, 0, stream>>>(g[0].W, Wt1, 4, F_);
  ktranspose<<<nb(F_ * F_, TB), TB, 0, stream>>>(g[1].W, Wt2, F_, F_);
  ktranspose<<<nb(F_ * F_, TB), TB, 0, stream>>>(g[2].W, Wt3, F_, F_);

  kzero<<<nb(B_ * A_ * HID_, TB), TB, 0, stream>>>(hst, B_ * A_ * HID_);

  for (int t = 0; t < T_; ++t) {
    const float* xt  = x_all + (size_t)t * NODES_ * 4;
    const int*   src = ei_all + (size_t)t * 2 * E_;
    const int*   dst = src + E_;
    const float* ea  = ea_all + (size_t)t * E_;

    // self-loop attrs (shared by GAT2 & GAT3)
    kzero<<<nb(NODES_, TB), TB, 0, stream>>>(sums, NODES_);
    kzero<<<nb(NODES_, TB), TB, 0, stream>>>(cnt, NODES_);
    kloop_accum<<<nb(E_, TB), TB, 0, stream>>>(dst, ea, sums, cnt);
    kloop_attr<<<nb(NODES_, TB), TB, 0, stream>>>(sums, cnt, la);

    for (int l = 0; l < 3; ++l) {
      const float* yin = (l == 0) ? xt : bufA;
      bool sl = (l != 0);

      kce<<<1, HEADS_, 0, stream>>>(g[l].lin, g[l].aedge, ce);
      {
        int blocks = ((NODES_ / 16) * (F_ / 64)) / 4;   // 1024
        if (l == 0)
          kgemmT<4><<<blocks, 128, 0, stream>>>(yin, Wt[0], bufB, NODES_, F_);
        else
          kgemmT<F_><<<blocks, 128, 0, stream>>>(yin, Wt[l], bufB, NODES_, F_);
      }
      kattn_coef<<<nb(NODES_ * HEADS_, TB), TB, 0, stream>>>(bufB, g[l].asrc, g[l].adst,
                                                             a_src, a_dst);
      kzero<<<nb((long)NODES_ * F_, TB), TB, 0, stream>>>(bufA, NODES_ * F_);
      kzero<<<nb(NODES_ * HEADS_, TB), TB, 0, stream>>>(denom, NODES_ * HEADS_);
      kinit_mkey<<<nb(NODES_ * HEADS_, TB), TB, 0, stream>>>(mkey, NODES_ * HEADS_);

      kedge_max<<<nb((long)E_ * HEADS_, TB), TB, 0, stream>>>(src, dst, ea, a_src, a_dst,
                                                              ce, alpha, mkey);
      if (sl)
        kself_max<<<nb(NODES_ * HEADS_, TB), TB, 0, stream>>>(la, a_src, a_dst, ce,
                                                              alphaS, mkey);
      kedge_sum<<<nb((long)E_ * HEADS_, TB), TB, 0, stream>>>(dst, alpha, mkey, denom);
      if (sl)
        kself_sum<<<nb(NODES_ * HEADS_, TB), TB, 0, stream>>>(alphaS, mkey, denom);

      kedge_agg<<<E_, F_, 0, stream>>>(src, dst, alpha, denom, bufB, bufA);
      if (sl)
        kself_agg<<<NODES_, F_, 0, stream>>>(alphaS, denom, bufB, bufA);

      kfinalize<<<nb((long)NODES_ * F_, TB), TB, 0, stream>>>(bufA, g[l].bias);
    }

    // readout + GRU + head
    kpool<<<nb(B_ * F_, TB), TB, 0, stream>>>(bufA, ge);
    kcomb<<<nb(B_ * A_ * GRUIN_, TB), TB, 0, stream>>>(bufA, ge, comb);
    {
      int blocks = ((B_ * A_ / 16) * (G3_ / 64) + 3) / 4;  // 120 tiles -> 30 blocks
      kgemmT<GRUIN_><<<blocks, 128, 0, stream>>>(comb, w_ih, gi, B_ * A_, G3_);
      kgemmT<HID_><<<blocks, 128, 0, stream>>>(hst, w_hh, gh, B_ * A_, G3_);
    }
    kgru_combine<<<nb(B_ * A_ * HID_, TB), TB, 0, stream>>>(gi, gh, b_ih, b_hh, hst);
    khead<<<nb(B_ * A_, TB), TB, 0, stream>>>(hst, fc1w, fc1b, fc2w, fc2b, out, t);
  }
}